// EGAMultiHeadedAttention2_90151363543417
// MI455X (gfx1250) — compile-verified
//
#include <hip/hip_runtime.h>
#include <hip/hip_bf16.h>
#include <math.h>

// ---------------------------------------------------------------------------
// EGA multi-headed attention for MI455X (gfx1250), fp32 via V_WMMA_F32_16X16X4_F32
// B=4, L=2048, E=512, H=8, Ek=Ev=64, Lk=32, Gk=16, Gq=8, Sk=1
// ---------------------------------------------------------------------------

typedef __attribute__((ext_vector_type(2))) float v2f;
typedef __attribute__((ext_vector_type(8))) float v8f;

#define CB 4
#define CL 2048
#define CE 512
#define CH 8
#define CEK 64
#define CLK 32
#define CGK 16
#define CGQ 8

__device__ __forceinline__ v8f wmma4(v2f a, v2f b, v8f c) {
  return __builtin_amdgcn_wmma_f32_16x16x4_f32(
      /*neg_a=*/false, a, /*neg_b=*/false, b,
      /*c_mod=*/(short)0, c, /*reuse_a=*/false, /*reuse_b=*/false);
}

__device__ __forceinline__ int iclamp(int v, int lo, int hi) {
  return v < lo ? lo : (v > hi ? hi : v);
}

// ---------------------------------------------------------------------------
// Batched strided fp32 WMMA GEMM: C = alpha * A(MxK) * op(B)(KxN).
// One wave computes a 32x16 C block (two 16x16 tiles sharing the B fragment),
// K stepped by 4. Row indices clamped (OOB rows never stored), so the inner
// loop has zero EXEC divergence: pure global_load_b64/b32 + v_wmma.
// ---------------------------------------------------------------------------
template <int TRANSB>
__global__ __launch_bounds__(128) void gemm_wmma_f32(
    const float* __restrict__ Ab, const float* __restrict__ Bb,
    float* __restrict__ Cb, int M, int N, int K, int lda, int ldb, int ldc,
    long sAb, long sAh, long sBb, long sBh, long sCb, long sCh, int Hdim,
    float alpha) {
  const int z = blockIdx.z;
  const int bb = z / Hdim, hh = z % Hdim;
  const float* A = Ab + (size_t)bb * sAb + (size_t)hh * sAh;
  const float* Bm = Bb + (size_t)bb * sBb + (size_t)hh * sBh;
  float* C = Cb + (size_t)bb * sCb + (size_t)hh * sCh;

  const int wave = threadIdx.x >> 5;
  const int lane = threadIdx.x & 31;
  const int tn = (blockIdx.x * 4 + wave) * 16;
  const int tm = blockIdx.y * 32;
  if (tn >= N) return;  // wave-uniform

  const int lm = lane & 15;
  const int kh = (lane >> 4) << 1;  // K sub-offset: 0 or 2
  const int ncol = tn + lm;
  const int m0 = iclamp(tm + lm, 0, M - 1);
  const int m1 = iclamp(tm + 16 + lm, 0, M - 1);

  const float* ap0 = A + (size_t)m0 * lda + kh;
  const float* ap1 = A + (size_t)m1 * lda + kh;
  const float* bp = TRANSB ? (Bm + (size_t)ncol * ldb + kh)
                           : (Bm + (size_t)kh * ldb + ncol);

  __builtin_prefetch(ap0, 0, 3);
  __builtin_prefetch(bp, 0, 3);

  v8f acc0 = {0.f, 0.f, 0.f, 0.f, 0.f, 0.f, 0.f, 0.f};
  v8f acc1 = {0.f, 0.f, 0.f, 0.f, 0.f, 0.f, 0.f, 0.f};

#pragma unroll 4
  for (int k0 = 0; k0 < K; k0 += 4) {
    const v2f a0 = *(const v2f*)ap0;
    const v2f a1 = *(const v2f*)ap1;
    v2f bv;
    if (TRANSB) {
      bv = *(const v2f*)bp;
      bp += 4;
    } else {
      bv.x = bp[0];
      bv.y = bp[ldb];
      bp += (size_t)ldb * 4;
    }
    ap0 += 4;
    ap1 += 4;
    acc0 = wmma4(a0, bv, acc0);
    acc1 = wmma4(a1, bv, acc1);
  }

  const int r0 = (lane >> 4) << 3;  // row base 0 or 8 within a tile
#pragma unroll
  for (int r = 0; r < 8; ++r) {
    const int mm0 = tm + r0 + r;
    const int mm1 = tm + 16 + r0 + r;
    if (mm0 < M) C[(size_t)mm0 * ldc + tn + lm] = alpha * acc0[r];
    if (mm1 < M) C[(size_t)mm1 * ldc + tn + lm] = alpha * acc1[r];
  }
}

// ---------------------------------------------------------------------------
// Softmax over the sequence dim for importance projections.
// raw: (B, L, ncols) -> out: (B*ncols, L)  with out row z = (b*ncols + col)
// ---------------------------------------------------------------------------
__global__ __launch_bounds__(256) void colsoftmax(const float* __restrict__ raw,
                                                  float* __restrict__ out,
                                                  int L, int ncols,
                                                  float scale) {
  const int z = blockIdx.x;
  const int b = z / ncols, col = z % ncols;
  const float* src = raw + (size_t)b * L * ncols + col;
  float* dst = out + (size_t)z * L;
  __shared__ float red[256];
  const int tid = threadIdx.x;

  float mx = -INFINITY;
  for (int i = tid; i < L; i += 256)
    mx = fmaxf(mx, src[(size_t)i * ncols] * scale);
  red[tid] = mx;
  __syncthreads();
  for (int s = 128; s > 0; s >>= 1) {
    if (tid < s) red[tid] = fmaxf(red[tid], red[tid + s]);
    __syncthreads();
  }
  mx = red[0];
  __syncthreads();

  float sum = 0.f;
  for (int i = tid; i < L; i += 256)
    sum += __expf(src[(size_t)i * ncols] * scale - mx);
  red[tid] = sum;
  __syncthreads();
  for (int s = 128; s > 0; s >>= 1) {
    if (tid < s) red[tid] += red[tid + s];
    __syncthreads();
  }
  const float inv = 1.f / red[0];
  for (int i = tid; i < L; i += 256)
    dst[i] = __expf(src[(size_t)i * ncols] * scale - mx) * inv;
}

// ---------------------------------------------------------------------------
// Fused attention: one wave handles 16 queries of one (b,h).
// Logits: local window as 3 WMMA tiles over the 48-key union window,
//         [gk(16) | gq(8)+static(1)] as 2 more tiles. 57-way softmax in LDS.
// Output: weights expanded to A(16x80), one WMMA GEMM vs V(80x64); V rows
//         48..79 staged in LDS. All loads clamped: no EXEC divergence.
// ---------------------------------------------------------------------------
__global__ __launch_bounds__(32) void ega_attn(
    const float* __restrict__ qs, const float* __restrict__ ks,
    const float* __restrict__ vs, const float* __restrict__ gkeys,
    const float* __restrict__ gkvals, const float* __restrict__ gqkeys,
    const float* __restrict__ gqvals, const float* __restrict__ stat_k,
    const float* __restrict__ stat_v, float* __restrict__ ao) {
  const int q0 = blockIdx.x * 16;
  const int z = blockIdx.y;
  const int b = z / CH, h = z % CH;
  const int RS = CH * CEK;  // 512: row stride of (B,L,H,Ek) layouts

  const float* qbase = qs + ((size_t)b * CL * CH + h) * CEK;
  const float* kbase = ks + ((size_t)b * CL * CH + h) * CEK;
  const float* vbase = vs + ((size_t)b * CL * CH + h) * CEK;
  const float* gk = gkeys + (size_t)z * CGK * CEK;
  const float* gkv = gkvals + (size_t)z * CGK * CEK;
  const float* gq = gqkeys + (size_t)z * CGQ * CEK;
  const float* gqv = gqvals + (size_t)z * CGQ * CEK;
  const float* sk = stat_k + (size_t)h * CEK;
  const float* sv = stat_v + (size_t)h * CEK;

  const int lane = threadIdx.x & 31;
  const int lm = lane & 15;
  const int kh = (lane >> 4) << 1;
  const int r0 = (lane >> 4) << 3;

  __shared__ float slog[16][80];  // cols: 0..47 local union, 48..63 gk,
  __shared__ float sA[16][80];    //       64..71 gq, 72 static, 73..79 pad
  __shared__ float sV[32][64];    // V rows 48..79 (gk|gq|static|zero)

  // ---- stage small V rows into LDS (uniform branches per chunk) ----------
  for (int i = lane; i < 32 * 64; i += 32) {
    const int row = i >> 6, col = i & 63;
    float v;
    if (row < 16)      v = gkv[row * CEK + col];
    else if (row < 24) v = gqv[(row - 16) * CEK + col];
    else if (row == 24) v = sv[col];
    else               v = 0.f;
    sV[row][col] = v;
  }

  // ---- logits via WMMA: per-lane clamped base pointers, then clean loop --
  const float* qp = qbase + (size_t)(q0 + lm) * RS + kh;
  const float* kp0;
  const float* kp1;
  const float* kp2;
  {
    const int p0 = iclamp(q0 - 16 + lm, 0, CL - 1);
    const int p1 = iclamp(q0 + lm, 0, CL - 1);
    const int p2 = iclamp(q0 + 16 + lm, 0, CL - 1);
    kp0 = kbase + (size_t)p0 * RS + kh;
    kp1 = kbase + (size_t)p1 * RS + kh;
    kp2 = kbase + (size_t)p2 * RS + kh;
  }
  const float* gkp = gk + lm * CEK + kh;
  // lanes 0..7 -> gq rows, lanes 8..15 -> static key (cols 9..15 unused)
  const float* sqp = (lm < 8) ? (gq + lm * CEK + kh) : (sk + kh);

  __builtin_prefetch(qp, 0, 3);
  __builtin_prefetch(kp1, 0, 3);

  v8f acc0 = {0.f, 0.f, 0.f, 0.f, 0.f, 0.f, 0.f, 0.f};
  v8f acc1 = acc0, acc2 = acc0, acc3 = acc0, acc4 = acc0;

#pragma unroll 4
  for (int k0 = 0; k0 < CEK; k0 += 4) {
    const v2f a = *(const v2f*)qp;  qp += 4;
    const v2f b0 = *(const v2f*)kp0; kp0 += 4;
    const v2f b1 = *(const v2f*)kp1; kp1 += 4;
    const v2f b2 = *(const v2f*)kp2; kp2 += 4;
    const v2f b3 = *(const v2f*)gkp; gkp += 4;
    const v2f b4 = *(const v2f*)sqp; sqp += 4;
    acc0 = wmma4(a, b0, acc0);
    acc1 = wmma4(a, b1, acc1);
    acc2 = wmma4(a, b2, acc2);
    acc3 = wmma4(a, b3, acc3);
    acc4 = wmma4(a, b4, acc4);
  }

#pragma unroll
  for (int r = 0; r < 8; ++r) {
    const int m = r0 + r;
    slog[m][lm] = acc0[r];
    slog[m][16 + lm] = acc1[r];
    slog[m][32 + lm] = acc2[r];
    slog[m][48 + lm] = acc3[r];
    slog[m][64 + lm] = acc4[r];
  }
  __syncthreads();

  // ---- per-row 57-way softmax (lanes 0..15, one query row each) ----------
  if (lane < 16) {
    const int m = lane;
    float mx = -INFINITY;
    for (int j = 0; j < CLK; ++j) {
      const int c = m + j;
      const int p = q0 - 16 + c;
      if (p >= 0 && p < CL) mx = fmaxf(mx, slog[m][c]);
    }
    for (int c = 48; c < 73; ++c) mx = fmaxf(mx, slog[m][c]);
    for (int c = 0; c < 80; ++c) sA[m][c] = 0.f;
    float sum = 0.f;
    for (int j = 0; j < CLK; ++j) {
      const int c = m + j;
      const int p = q0 - 16 + c;
      if (p >= 0 && p < CL) {
        const float e = __expf(slog[m][c] - mx);
        sum += e;
        sA[m][c] = e;
      }
    }
    for (int c = 48; c < 73; ++c) {
      const float e = __expf(slog[m][c] - mx);
      sum += e;
      sA[m][c] = e;
    }
    const float inv = 1.f / sum;
    for (int c = 0; c < 80; ++c) sA[m][c] *= inv;
  }
  __syncthreads();

  // ---- output: A(16x80) @ V(80x64) via WMMA ------------------------------
  v8f o0 = {0.f, 0.f, 0.f, 0.f, 0.f, 0.f, 0.f, 0.f};
  v8f o1 = o0, o2 = o0, o3 = o0;

  for (int k0 = 0; k0 < 80; k0 += 4) {
    const int k = k0 + kh;
    const v2f a = *(const v2f*)&sA[lm][k];
    v2f b0, b1, b2, b3;
    if (k0 < 48) {  // window values (uniform branch: region is 4-aligned)
      const int p0 = iclamp(q0 - 16 + k, 0, CL - 1);
      const int p1 = iclamp(q0 - 15 + k, 0, CL - 1);
      const float* v0 = vbase + (size_t)p0 * RS;
      const float* v1 = vbase + (size_t)p1 * RS;
      b0.x = v0[lm];      b0.y = v1[lm];
      b1.x = v0[16 + lm]; b1.y = v1[16 + lm];
      b2.x = v0[32 + lm]; b2.y = v1[32 + lm];
      b3.x = v0[48 + lm]; b3.y = v1[48 + lm];
    } else {  // gk/gq/static values from LDS
      const int r = k - 48;
      b0.x = sV[r][lm];      b0.y = sV[r + 1][lm];
      b1.x = sV[r][16 + lm]; b1.y = sV[r + 1][16 + lm];
      b2.x = sV[r][32 + lm]; b2.y = sV[r + 1][32 + lm];
      b3.x = sV[r][48 + lm]; b3.y = sV[r + 1][48 + lm];
    }
    o0 = wmma4(a, b0, o0);
    o1 = wmma4(a, b1, o1);
    o2 = wmma4(a, b2, o2);
    o3 = wmma4(a, b3, o3);
  }

  float* obase = ao + ((size_t)b * CL * CH + h) * CEK;
#pragma unroll
  for (int r = 0; r < 8; ++r) {
    const int l = q0 + r0 + r;
    obase[(size_t)l * RS + lm] = o0[r];
    obase[(size_t)l * RS + 16 + lm] = o1[r];
    obase[(size_t)l * RS + 32 + lm] = o2[r];
    obase[(size_t)l * RS + 48 + lm] = o3[r];
  }
}

// ---------------------------------------------------------------------------
// Host side
// ---------------------------------------------------------------------------
static inline void launch_gemm(hipStream_t s, const float* A, const float* B,
                               float* C, int M, int N, int K, int lda, int ldb,
                               int ldc, long sAb, long sAh, long sBb, long sBh,
                               long sCb, long sCh, int Bdim, int Hdim,
                               int transB, float alpha) {
  dim3 grid((N + 63) / 64, (M + 31) / 32, Bdim * Hdim);
  if (transB)
    gemm_wmma_f32<1><<<grid, dim3(128), 0, s>>>(A, B, C, M, N, K, lda, ldb,
                                                ldc, sAb, sAh, sBb, sBh, sCb,
                                                sCh, Hdim, alpha);
  else
    gemm_wmma_f32<0><<<grid, dim3(128), 0, s>>>(A, B, C, M, N, K, lda, ldb,
                                                ldc, sAb, sAh, sBb, sBh, sCb,
                                                sCh, Hdim, alpha);
}

extern "C" void kernel_launch(void* const* d_in, const int* in_sizes, int n_in,
                              void* d_out, int out_size, void* d_ws,
                              size_t ws_size, hipStream_t stream) {
  const float* query = (const float*)d_in[0];
  const float* key = (const float*)d_in[1];
  const float* value = (const float*)d_in[2];
  const float* qproj = (const float*)d_in[4];
  const float* kproj = (const float*)d_in[5];
  const float* vproj = (const float*)d_in[6];
  const float* kiproj = (const float*)d_in[7];
  const float* qiproj = (const float*)d_in[8];
  const float* stat_k = (const float*)d_in[9];
  const float* stat_v = (const float*)d_in[10];
  const float* oproj = (const float*)d_in[11];
  float* out = (float*)d_out;

  const int Bm = CB, L = CL, E = CE, H = CH, EK = CEK;
  const int M = Bm * L;  // 8192

  float* w = (float*)d_ws;
  size_t o = 0;
  float* qs = w + o;        o += (size_t)M * E;            // (B,L,H,Ek)
  float* ks = w + o;        o += (size_t)M * E;
  float* vsb = w + o;       o += (size_t)M * E;
  float* kimp_raw = w + o;  o += (size_t)M * H * CGK;      // (B,L,128)
  float* qimp_raw = w + o;  o += (size_t)M * H * CGQ;      // (B,L,64)
  float* key_imps = w + o;  o += (size_t)Bm * H * CGK * L; // (B,H,Gk,L)
  float* query_imps = w + o;o += (size_t)Bm * H * CGQ * L; // (B,H,Gq,L)
  float* gkeys = w + o;     o += (size_t)Bm * H * CGK * EK;
  float* gkvals = w + o;    o += (size_t)Bm * H * CGK * EK;
  float* gquerys = w + o;   o += (size_t)Bm * H * CGQ * EK;
  float* inter = w + o;     o += (size_t)Bm * H * CGQ * L; // (B,H,Gq,L)
  float* gqkeys = w + o;    o += (size_t)Bm * H * CGQ * EK;
  float* gqvals = w + o;    o += (size_t)Bm * H * CGQ * EK;
  float* ao = w + o;        o += (size_t)M * E;            // (B,L,H,Ev)

  const float iscale = 1.0f / sqrtf((float)E);   // 1/sqrt(512)
  const float ekscale = 1.0f / sqrtf((float)EK); // 1/8

  // 1-5: projections (batch 1)
  launch_gemm(stream, query, qproj, qs, M, E, E, E, E, E, 0, 0, 0, 0, 0, 0, 1,
              1, 0, 1.f);
  launch_gemm(stream, key, kproj, ks, M, E, E, E, E, E, 0, 0, 0, 0, 0, 0, 1, 1,
              0, 1.f);
  launch_gemm(stream, value, vproj, vsb, M, E, E, E, E, E, 0, 0, 0, 0, 0, 0, 1,
              1, 0, 1.f);
  launch_gemm(stream, key, kiproj, kimp_raw, M, H * CGK, E, E, H * CGK,
              H * CGK, 0, 0, 0, 0, 0, 0, 1, 1, 0, 1.f);
  launch_gemm(stream, query, qiproj, qimp_raw, M, H * CGQ, E, E, H * CGQ,
              H * CGQ, 0, 0, 0, 0, 0, 0, 1, 1, 0, 1.f);

  // 6-7: importance softmaxes over L
  colsoftmax<<<dim3(Bm * H * CGK), dim3(256), 0, stream>>>(
      kimp_raw, key_imps, L, H * CGK, iscale);
  colsoftmax<<<dim3(Bm * H * CGQ), dim3(256), 0, stream>>>(
      qimp_raw, query_imps, L, H * CGQ, iscale);

  // 8-9: global_keys / gk_values = key_imps @ {ks, vs}  (per (b,h))
  launch_gemm(stream, key_imps, ks, gkeys, CGK, EK, L, L, H * EK, EK,
              (long)H * CGK * L, (long)CGK * L, (long)L * H * EK, EK,
              (long)H * CGK * EK, (long)CGK * EK, Bm, H, 0, 1.f);
  launch_gemm(stream, key_imps, vsb, gkvals, CGK, EK, L, L, H * EK, EK,
              (long)H * CGK * L, (long)CGK * L, (long)L * H * EK, EK,
              (long)H * CGK * EK, (long)CGK * EK, Bm, H, 0, 1.f);

  // 10: global_querys = query_imps @ qs
  launch_gemm(stream, query_imps, qs, gquerys, CGQ, EK, L, L, H * EK, EK,
              (long)H * CGQ * L, (long)CGQ * L, (long)L * H * EK, EK,
              (long)H * CGQ * EK, (long)CGQ * EK, Bm, H, 0, 1.f);

  // 11: inter = global_querys @ ks^T / sqrt(Ek)
  launch_gemm(stream, gquerys, ks, inter, CGQ, L, EK, EK, H * EK, L,
              (long)H * CGQ * EK, (long)CGQ * EK, (long)L * H * EK, EK,
              (long)H * CGQ * L, (long)CGQ * L, Bm, H, 1, ekscale);

  // 12-13: gq_keys / gq_values = inter @ {ks, vs}
  launch_gemm(stream, inter, ks, gqkeys, CGQ, EK, L, L, H * EK, EK,
              (long)H * CGQ * L, (long)CGQ * L, (long)L * H * EK, EK,
              (long)H * CGQ * EK, (long)CGQ * EK, Bm, H, 0, 1.f);
  launch_gemm(stream, inter, vsb, gqvals, CGQ, EK, L, L, H * EK, EK,
              (long)H * CGQ * L, (long)CGQ * L, (long)L * H * EK, EK,
              (long)H * CGQ * EK, (long)CGQ * EK, Bm, H, 0, 1.f);

  // 14: fused local + global attention
  ega_attn<<<dim3(L / 16, Bm * H), dim3(32), 0, stream>>>(
      qs, ks, vsb, gkeys, gkvals, gqkeys, gqvals, stat_k, stat_v, ao);

  // 15: output projection -> d_out
  launch_gemm(stream, ao, oproj, out, M, E, E, E, E, E, 0, 0, 0, 0, 0, 0, 1, 1,
              0, 1.f);

  (void)in_sizes; (void)n_in; (void)out_size; (void)ws_size;
}